// DeformableEncoderLayer_12463995093369
// MI455X (gfx1250) — compile-verified
//
#include <hip/hip_runtime.h>
#include <hip/hip_bf16.h>

typedef __attribute__((ext_vector_type(2))) float v2f;
typedef __attribute__((ext_vector_type(8))) float v8f;

#define N_TOTAL 21760

// ---------------------------------------------------------------------------
// LayerNorm over D=256. One block (256 threads) per row.
// outv = LN(x); if outq: outq = outv + pos.
// ---------------------------------------------------------------------------
__global__ __launch_bounds__(256) void ln_kernel(
    const float* __restrict__ x, const float* __restrict__ w,
    const float* __restrict__ b, const float* __restrict__ pos,
    float* __restrict__ outv, float* __restrict__ outq)
{
  const int row = blockIdx.x;
  const int t = threadIdx.x;
  const size_t base = (size_t)row * 256 + t;
  float xv = x[base];
  float s = xv, s2 = xv * xv;
  #pragma unroll
  for (int o = 16; o > 0; o >>= 1) {
    s  += __shfl_xor(s,  o, 32);
    s2 += __shfl_xor(s2, o, 32);
  }
  __shared__ float rs[8], rs2[8];
  const int wave = t >> 5, lane = t & 31;
  if (lane == 0) { rs[wave] = s; rs2[wave] = s2; }
  __syncthreads();
  if (wave == 0) {
    float a  = (lane < 8) ? rs[lane]  : 0.f;
    float a2 = (lane < 8) ? rs2[lane] : 0.f;
    #pragma unroll
    for (int o = 4; o > 0; o >>= 1) {
      a += __shfl_xor(a, o, 32); a2 += __shfl_xor(a2, o, 32);
    }
    if (lane == 0) { rs[0] = a; rs2[0] = a2; }
  }
  __syncthreads();
  const float mean = rs[0] * (1.f / 256.f);
  const float var  = rs2[0] * (1.f / 256.f) - mean * mean;
  const float r = rsqrtf(var + 1e-5f);
  const float v = (xv - mean) * r * w[t] + b[t];
  outv[base] = v;
  if (outq) outq[base] = v + pos[base];
}

// ---------------------------------------------------------------------------
// Generic fp32 WMMA GEMM: C[M,N] = op(A[M,K] @ W[K,N] + bias [+ resid])
// Block = 256 threads (8 waves). Each block owns a 16-row tile of A (staged
// in LDS, row stride K+4 -> conflict-free). Wave w computes the 16x16 output
// tile at columns (blockIdx.y*8 + w)*16 via V_WMMA_F32_16X16X4_F32, K/4 steps.
//
// A-fragment layout (ISA 7.12.2, 32-bit A 16x4): lanes 0-15 hold K={0,1},
// lanes 16-31 hold K={2,3}; row M = lane&15.
// B-fragment (4x16, rows striped across lanes within a VGPR):
// VGPR j: lanes 0-15 -> K=j, lanes 16-31 -> K=j+2; column N = lane&15.
// C/D layout: VGPR j: lanes 0-15 -> M=j, lanes 16-31 -> M=j+8; N = lane&15.
// ---------------------------------------------------------------------------
template<bool RELU, bool RESID>
__global__ __launch_bounds__(256) void gemm_wmma_f32(
    const float* __restrict__ A, const float* __restrict__ W,
    const float* __restrict__ bias, const float* __restrict__ resid,
    float* __restrict__ C, int M, int K, int N, int kshift)
{
  extern __shared__ float lds[];
  const int tid = threadIdx.x;
  const int m0 = blockIdx.x * 16;
  const int Kp = K + 4;

  // cooperative stage of the 16xK A tile (coalesced global reads)
  for (int i = tid; i < 16 * K; i += 256) {
    const int r = i >> kshift;        // K is a power of two
    const int c = i & (K - 1);
    lds[r * Kp + c] = A[(size_t)(m0 + r) * K + c];
  }
  __syncthreads();

  const int wave = tid >> 5, lane = tid & 31;
  const int n0 = (blockIdx.y * 8 + wave) * 16;
  if (n0 >= N) return;

  const int mrow = lane & 15;
  const int kb   = (lane >> 4) << 1;          // 0 for lanes 0-15, 2 for 16-31
  const int ncol = n0 + (lane & 15);

  v8f acc = {};
  const float* __restrict__ lrow = lds + mrow * Kp + kb;
  for (int k = 0; k < K; k += 4) {
    v2f a; a.x = lrow[k];     a.y = lrow[k + 1];
    v2f bfr;
    bfr.x = W[(size_t)(k + kb)     * N + ncol];
    bfr.y = W[(size_t)(k + kb + 1) * N + ncol];
    acc = __builtin_amdgcn_wmma_f32_16x16x4_f32(
        /*neg_a=*/false, a, /*neg_b=*/false, bfr,
        /*c_mod=*/(short)0, acc, /*reuse_a=*/false, /*reuse_b=*/false);
  }

  const float bval = bias[ncol];
  const int rbase = m0 + ((lane >> 4) << 3);
  #pragma unroll
  for (int j = 0; j < 8; ++j) {
    const size_t o = (size_t)(rbase + j) * N + ncol;
    float v = acc[j] + bval;
    if (RELU)  v = fmaxf(v, 0.f);
    if (RESID) v += resid[o];
    C[o] = v;
  }
}

// ---------------------------------------------------------------------------
// MSDA sampling core. One wave per (b, n, h); lane = channel c (HD = 32).
// Softmax over the 16 (L*P) logits is computed lane-redundantly (broadcast
// loads); every bilinear gather is a coalesced 128B line (value tensor is
// L2-resident: 45 MB << 192 MB L2).
// loc_x * W_l - 0.5  ==  ref_x * W_l + off_x - 0.5  (norm cancels).
// ---------------------------------------------------------------------------
__global__ __launch_bounds__(256) void msda_sample_kernel(
    const float* __restrict__ value,  // (B*N, 256) = (B*N, H, 32)
    const float* __restrict__ offs,   // (B*N, 256) = (B*N, H, L, P, 2)
    const float* __restrict__ logits, // (B*N, 128) = (B*N, H, L*P)
    const float* __restrict__ ref,    // (B*N, L, 2)
    float* __restrict__ out)          // (B*N, 256)
{
  const int wid  = blockIdx.x * 8 + (threadIdx.x >> 5);
  const int lane = threadIdx.x & 31;
  const int h  = wid & 7;
  const int bn = wid >> 3;
  const int bidx = bn / N_TOTAL;
  const size_t vrowbase = (size_t)bidx * N_TOTAL;

  // softmax over 16 logits
  const float* lp = logits + (size_t)bn * 128 + h * 16;
  float lg[16];
  float mx = -3.0e38f;
  #pragma unroll
  for (int i = 0; i < 16; ++i) { lg[i] = lp[i]; mx = fmaxf(mx, lg[i]); }
  float ssum = 0.f;
  #pragma unroll
  for (int i = 0; i < 16; ++i) { lg[i] = __expf(lg[i] - mx); ssum += lg[i]; }
  const float inv = 1.f / ssum;

  const float* op = offs + (size_t)bn * 256 + h * 32;  // (L,P,2) for this head
  const float* rp = ref  + (size_t)bn * 8;             // (L,2)

  const int starts[4] = {0, 16384, 20480, 21504};
  const int dims[4]   = {128, 64, 32, 16};             // square levels

  float acc = 0.f;
  #pragma unroll
  for (int l = 0; l < 4; ++l) {
    const int Wl = dims[l], Hl = dims[l];
    const float fW = (float)Wl, fH = (float)Hl;
    const float rx = rp[l * 2], ry = rp[l * 2 + 1];
    #pragma unroll
    for (int p = 0; p < 4; ++p) {
      const float aw = lg[l * 4 + p] * inv;
      const float ox = op[(l * 4 + p) * 2];
      const float oy = op[(l * 4 + p) * 2 + 1];
      const float xx = rx * fW + ox - 0.5f;
      const float yy = ry * fH + oy - 0.5f;
      const float x0f = floorf(xx), y0f = floorf(yy);
      const float wx = xx - x0f, wy = yy - y0f;
      const int x0 = (int)x0f, y0 = (int)y0f;
      #pragma unroll
      for (int corner = 0; corner < 4; ++corner) {
        const int dx = corner & 1, dy = corner >> 1;
        const int xi = x0 + dx, yi = y0 + dy;
        const bool valid = (xi >= 0) & (xi < Wl) & (yi >= 0) & (yi < Hl);
        const float cw = (dx ? wx : 1.f - wx) * (dy ? wy : 1.f - wy);
        const int xc = min(max(xi, 0), Wl - 1);
        const int yc = min(max(yi, 0), Hl - 1);
        const size_t vrow = vrowbase + starts[l] + (size_t)yc * Wl + xc;
        const float g = value[vrow * 256 + h * 32 + lane];
        acc += (valid ? cw * aw : 0.f) * g;
      }
    }
  }
  out[(size_t)bn * 256 + h * 32 + lane] = acc;
}

// ---------------------------------------------------------------------------
extern "C" void kernel_launch(void* const* d_in, const int* in_sizes, int n_in,
                              void* d_out, int out_size, void* d_ws, size_t ws_size,
                              hipStream_t stream) {
  const float* embed = (const float*)d_in[0];
  const float* pos   = (const float*)d_in[1];
  const float* refp  = (const float*)d_in[2];
  const float* ln1w  = (const float*)d_in[3];
  const float* ln1b  = (const float*)d_in[4];
  const float* ln2w  = (const float*)d_in[5];
  const float* ln2b  = (const float*)d_in[6];
  const float* Wv    = (const float*)d_in[7];
  const float* bv    = (const float*)d_in[8];
  const float* Woff  = (const float*)d_in[9];
  const float* boff  = (const float*)d_in[10];
  const float* Wattn = (const float*)d_in[11];
  const float* battn = (const float*)d_in[12];
  const float* Wo    = (const float*)d_in[13];
  const float* bo    = (const float*)d_in[14];
  const float* W1    = (const float*)d_in[15];
  const float* b1    = (const float*)d_in[16];
  const float* W2    = (const float*)d_in[17];
  const float* b2    = (const float*)d_in[18];
  float* outp = (float*)d_out;

  const int M = in_sizes[0] / 256;            // B * N_TOTAL = 43520
  const int MT = M / 16;                      // 2720 row tiles

  // workspace layout (floats), with lifetime-based aliasing:
  // q      [0,        M*256)   later reused as f
  // vbuf   [M*256,    2M*256)  later reused as msda_out
  // embed2 [2M*256,   3M*256)
  // value  [3M*256,   4M*256)  dead after sampling
  // offb   [4M*256,   5M*256)  dead after sampling
  // logit  [5M*256,   5M*256 + M*128) dead after sampling
  // h1 aliases [3M*256, 3M*256 + M*1024)
  float* q      = (float*)d_ws;
  float* vbuf   = q      + (size_t)M * 256;
  float* embed2 = vbuf   + (size_t)M * 256;
  float* value  = embed2 + (size_t)M * 256;
  float* offb   = value  + (size_t)M * 256;
  float* logit  = offb   + (size_t)M * 256;
  float* h1     = value;
  float* msda   = vbuf;
  float* f      = q;

  const size_t sh256  = (size_t)16 * (256 + 4)  * sizeof(float);
  const size_t sh1024 = (size_t)16 * (1024 + 4) * sizeof(float);

  // 1. LN1 -> v, q = v + pos
  ln_kernel<<<M, 256, 0, stream>>>(embed, ln1w, ln1b, pos, vbuf, q);

  // 2. value = v @ Wv + bv            (M x 256)
  gemm_wmma_f32<false, false><<<dim3(MT, 2), 256, sh256, stream>>>(
      vbuf, Wv, bv, nullptr, value, M, 256, 256, 8);

  // 3. off = q @ Woff + boff          (M x 256)
  gemm_wmma_f32<false, false><<<dim3(MT, 2), 256, sh256, stream>>>(
      q, Woff, boff, nullptr, offb, M, 256, 256, 8);

  // 4. logits = q @ Wattn + battn     (M x 128)
  gemm_wmma_f32<false, false><<<dim3(MT, 1), 256, sh256, stream>>>(
      q, Wattn, battn, nullptr, logit, M, 256, 128, 8);

  // 5. multiscale deformable sampling -> msda (M x 256)
  msda_sample_kernel<<<M, 256, 0, stream>>>(value, offb, logit, refp, msda);

  // 6. embed2 = embed + msda @ Wo + bo
  gemm_wmma_f32<false, true><<<dim3(MT, 2), 256, sh256, stream>>>(
      msda, Wo, bo, embed, embed2, M, 256, 256, 8);

  // 7. LN2 -> f
  ln_kernel<<<M, 256, 0, stream>>>(embed2, ln2w, ln2b, nullptr, f, nullptr);

  // 8. h1 = relu(f @ W1 + b1)         (M x 1024)
  gemm_wmma_f32<true, false><<<dim3(MT, 8), 256, sh256, stream>>>(
      f, W1, b1, nullptr, h1, M, 256, 1024, 8);

  // 9. out = embed2 + h1 @ W2 + b2    (M x 256, K = 1024)
  gemm_wmma_f32<false, true><<<dim3(MT, 2), 256, sh1024, stream>>>(
      h1, W2, b2, embed2, outp, M, 1024, 256, 10);
}